// GaussianEmbedding_49211735277727
// MI455X (gfx1250) — compile-verified
//
#include <hip/hip_runtime.h>

typedef __attribute__((ext_vector_type(2))) float v2f;
typedef __attribute__((ext_vector_type(8))) float v8f;

#define E_DIM 128
#define MARGIN_F 0.1f
#define PAD 128              // hinge buffer padded to multiple of 128 (= 2 WMMA chunks)

// Per-element contribution to KL(t||c) with the terms that cancel in
// (kl_pos - kl_neg) already dropped:  (s_t + d^2) / s_c + log(s_c)
__device__ __forceinline__ float kl_term(float m_t, float s_t, float m_c, float s_c) {
    float d = m_c - m_t;
    return fmaf(s_t + d * d, __frcp_rn(s_c), __logf(s_c));
}

// One wave32 per (b,w) pair. Lane l owns elements [4l, 4l+4) of E=128 via
// float4 (global_load_b128). Single combined wave reduction of
// (pos_terms - neg_terms); hinge written by lane 0. Waves past n write 0 so
// the WMMA reduction can run guard-free over the padded buffer.
__global__ void gauss_hinge_kernel(const float* __restrict__ mu,
                                   const float* __restrict__ sigma,
                                   const int* __restrict__ target,
                                   const int* __restrict__ cpos,
                                   const int* __restrict__ cneg,
                                   float* __restrict__ hinge,
                                   int B, int W, int n_pad) {
    const int gtid = blockIdx.x * blockDim.x + threadIdx.x;
    const int gw   = gtid >> 5;          // global wave id = (b,w) pair
    const int lane = gtid & 31;
    const int n    = B * W;
    if (gw >= n) {                       // zero-fill padding region
        if (gw < n_pad && lane == 0) hinge[gw] = 0.0f;
        return;
    }

    const int b = gw / W;
    const int w = gw - b * W;

    const int it = target[b];            // target is [B,1]
    const int ip = cpos[b * W + w];
    const int in = cneg[b * W + w];

    const float4* mt4 = reinterpret_cast<const float4*>(mu    + (size_t)it * E_DIM);
    const float4* st4 = reinterpret_cast<const float4*>(sigma + (size_t)it * E_DIM);
    const float4* mp4 = reinterpret_cast<const float4*>(mu    + (size_t)ip * E_DIM);
    const float4* sp4 = reinterpret_cast<const float4*>(sigma + (size_t)ip * E_DIM);
    const float4* mn4 = reinterpret_cast<const float4*>(mu    + (size_t)in * E_DIM);
    const float4* sn4 = reinterpret_cast<const float4*>(sigma + (size_t)in * E_DIM);

    float4 mt = mt4[lane], st = st4[lane];
    float4 mp = mp4[lane], sp = sp4[lane];
    float4 mn = mn4[lane], sn = sn4[lane];

    float acc = 0.0f;
    acc += kl_term(mt.x, st.x, mp.x, sp.x) - kl_term(mt.x, st.x, mn.x, sn.x);
    acc += kl_term(mt.y, st.y, mp.y, sp.y) - kl_term(mt.y, st.y, mn.y, sn.y);
    acc += kl_term(mt.z, st.z, mp.z, sp.z) - kl_term(mt.z, st.z, mn.z, sn.z);
    acc += kl_term(mt.w, st.w, mp.w, sp.w) - kl_term(mt.w, st.w, mn.w, sn.w);

    // wave32 butterfly reduction
    #pragma unroll
    for (int off = 16; off > 0; off >>= 1)
        acc += __shfl_xor(acc, off, 32);

    if (lane == 0)
        hinge[gw] = fmaxf(0.0f, MARGIN_F - 0.5f * acc);   // M - (klp - kln)
}

// Deterministic final reduction of n_pad (multiple of 128) hinge values on the
// f32 matrix unit: A = 16x4 chunk (2 VGPRs/lane), B = ones. Two independent
// accumulator chains hide the D->C WMMA latency; loads are unconditional
// global_load_b64 (buffer is zero-padded by kernel 1).
__global__ void reduce_wmma_kernel(const float* __restrict__ hinge,
                                   float* __restrict__ out,
                                   int n_pad, float invB) {
    const int lane = threadIdx.x;        // launched with exactly 32 threads

    v8f c0 = {}, c1 = {};
    v2f ones; ones.x = 1.0f; ones.y = 1.0f;

    // lane's v2f slot within a 64-value chunk (32 v2f per chunk):
    //   lanes 0-15  hold rows M=lane,    K=0,1
    //   lanes 16-31 hold rows M=lane-16, K=2,3
    const int slot  = (lane & 15) * 2 + (lane >> 4);
    const v2f* h2   = reinterpret_cast<const v2f*>(hinge);
    const int pairs = n_pad >> 7;        // two 64-chunks per iteration

    for (int it = 0; it < pairs; ++it) {
        v2f a0 = h2[it * 64 + slot];
        v2f a1 = h2[it * 64 + 32 + slot];
        c0 = __builtin_amdgcn_wmma_f32_16x16x4_f32(false, a0, false, ones,
                                                   (short)0, c0, false, false);
        c1 = __builtin_amdgcn_wmma_f32_16x16x4_f32(false, a1, false, ones,
                                                   (short)0, c1, false, false);
    }

    v8f c = c0 + c1;
    float s = c[0] + c[1] + c[2] + c[3] + c[4] + c[5] + c[6] + c[7];
    #pragma unroll
    for (int off = 16; off > 0; off >>= 1)
        s += __shfl_xor(s, off, 32);

    if (lane == 0)
        out[0] = s * (1.0f / 16.0f) * invB;  // columns replicated 16x; mean over B
}

extern "C" void kernel_launch(void* const* d_in, const int* in_sizes, int n_in,
                              void* d_out, int out_size, void* d_ws, size_t ws_size,
                              hipStream_t stream) {
    const float* mu    = (const float*)d_in[0];
    const float* sigma = (const float*)d_in[1];
    const int*   tgt   = (const int*)d_in[2];
    const int*   cpos  = (const int*)d_in[3];
    const int*   cneg  = (const int*)d_in[4];

    const int B = in_sizes[2];           // target is [B,1]
    const int W = in_sizes[3] / B;       // context is [B,W]
    const int n = B * W;
    const int n_pad = (n + PAD - 1) & ~(PAD - 1);

    float* hinge = (float*)d_ws;         // n_pad floats of scratch

    const int threads = 256;             // 8 wave32 per block
    const int blocks  = (n_pad * 32 + threads - 1) / threads;

    gauss_hinge_kernel<<<blocks, threads, 0, stream>>>(mu, sigma, tgt, cpos, cneg,
                                                       hinge, B, W, n_pad);
    reduce_wmma_kernel<<<1, 32, 0, stream>>>(hinge, (float*)d_out, n_pad,
                                             1.0f / (float)B);
}